// EfficientKAN_76519137346083
// MI455X (gfx1250) — compile-verified
//
#include <hip/hip_runtime.h>
#include <hip/hip_bf16.h>

// ---------------------------------------------------------------------------
// Problem constants (from the reference)
// ---------------------------------------------------------------------------
constexpr int BATCH     = 4096;
constexpr int DIM       = 2048;
constexpr int NB        = 12;               // NUM_KNOTS - DEGREE - 1
constexpr int OUT_COLS  = DIM * NB + DIM;   // 26624
constexpr int INTER_OFF = DIM * NB;         // 24576 (start of interaction cols)

// GEMM tiling
constexpr int BM = 128;
constexpr int BN = 64;
constexpr int BK = 32;

// LDS row strides in float2 units. Need stride*8 bytes ≡ 128 (mod 256) so the
// two 16-lane halves (which read adjacent K-pair rows) hit disjoint bank
// halves: stride ≡ 16 (mod 32).
constexpr int ASTR = BM + 16;   // 144
constexpr int BSTR = BN + 16;   // 80

typedef __attribute__((ext_vector_type(2))) float v2f;
typedef __attribute__((ext_vector_type(8))) float v8f;

// ---------------------------------------------------------------------------
// Kernel A: B-spline basis (degree 3, uniform knots on [-1,1]).
// One thread per (b, d); writes 12 floats (3x float4). Store-bandwidth bound.
// ---------------------------------------------------------------------------
__global__ __launch_bounds__(256) void kan_basis(const float* __restrict__ x,
                                                 float* __restrict__ out) {
    const int idx = blockIdx.x * blockDim.x + threadIdx.x;
    const int b = idx >> 11;          // / DIM
    const int d = idx & (DIM - 1);    // % DIM

    const float h  = 2.0f / 15.0f;
    const float t0 = -1.0f;

    float xc = fminf(fmaxf(x[idx], -1.0f), 1.0f);

    float N[15];
#pragma unroll
    for (int i = 0; i < 15; ++i) {
        float ti  = t0 + h * i;
        float ti1 = t0 + h * (i + 1);
        N[i] = (xc >= ti && xc < ti1) ? 1.0f : 0.0f;
    }
    N[14] = (xc >= t0 + h * 14.0f) ? 1.0f : 0.0f;   // reference's last-interval fix

#pragma unroll
    for (int p = 1; p <= 3; ++p) {
        const float inv = 1.0f / (h * (float)p);    // uniform knots: all denoms = p*h
#pragma unroll
        for (int i = 0; i < 14; ++i) {
            if (i < 15 - p) {
                float ti   = t0 + h * i;
                float tipp = t0 + h * (i + p + 1);
                float a  = (xc - ti) * inv;
                float bb = (tipp - xc) * inv;
                N[i] = a * N[i] + bb * N[i + 1];
            }
        }
    }

    float* o = out + (size_t)b * OUT_COLS + (size_t)d * NB;
    *(float4*)(o + 0) = make_float4(N[0], N[1], N[2],  N[3]);
    *(float4*)(o + 4) = make_float4(N[4], N[5], N[6],  N[7]);
    *(float4*)(o + 8) = make_float4(N[8], N[9], N[10], N[11]);
}

// ---------------------------------------------------------------------------
// Kernel B: S = sigmoid(W) into workspace
// ---------------------------------------------------------------------------
__global__ __launch_bounds__(256) void kan_sigmoid(const float* __restrict__ W,
                                                   float* __restrict__ S, int n) {
    int i = blockIdx.x * blockDim.x + threadIdx.x;
    if (i < n) S[i] = 1.0f / (1.0f + __expf(-W[i]));
}

// ---------------------------------------------------------------------------
// Kernel C: interaction GEMM  C = (x .* x) @ S, f32, V_WMMA_F32_16X16X4_F32.
// 256 threads = 8 waves; block tile 128x64; wave tile 32x32 (2x2 accumulators).
// LDS holds K-interleaved pairs: As2[k/2][m] = {A[k][m], A[k+1][m]}, so a
// lane's WMMA fragment (K = 2*half, 2*half+1 per ISA 7.12.2) is a single
// ds_load_b64 into a consecutive VGPR pair — no repacking movs.
// Double-buffered LDS, one barrier per K-block; global loads for block k+1
// are issued before computing block k. All control flow is wave-uniform, so
// EXEC is all-ones at every WMMA (ISA 7.12 requirement).
// ---------------------------------------------------------------------------
template <bool FUSE_SIGMOID>
__global__ __launch_bounds__(256) void kan_gemm(const float* __restrict__ X,
                                                const float* __restrict__ Bsrc,
                                                float* __restrict__ out) {
    __shared__ float2 As2[2][BK / 2][ASTR];
    __shared__ float2 Bs2[2][BK / 2][BSTR];

    const int t    = threadIdx.x;
    const int m0   = blockIdx.y * BM;
    const int n0   = blockIdx.x * BN;
    const int wid  = t >> 5;
    const int lane = t & 31;
    const int half = lane >> 4;
    const int lm   = lane & 15;
    const int wm   = (wid >> 1) * 32;  // wave row offset (0,32,64,96)
    const int wn   = (wid & 1) * 32;   // wave col offset (0,32)

    // per-thread staging coordinates
    const int a_row = t >> 3;          // base row for A loads (it adds 32 rows)
    const int a_kq  = t & 7;           // which float4 along K
    const int b_kp  = t >> 4;          // K-pair index 0..15
    const int b_nq  = t & 15;          // float4 index along N

    float4 ar[4];
    float4 br0, br1;

    auto load_global = [&](int kb) {
#pragma unroll
        for (int it = 0; it < 4; ++it) {
            int row = a_row + 32 * it;
            ar[it] = *(const float4*)(X + (size_t)(m0 + row) * DIM + kb + a_kq * 4);
        }
        const float* bp = Bsrc + (size_t)(kb + 2 * b_kp) * DIM + n0 + b_nq * 4;
        br0 = *(const float4*)bp;
        br1 = *(const float4*)(bp + DIM);
    };

    auto store_lds = [&](int buf) {
#pragma unroll
        for (int it = 0; it < 4; ++it) {
            int row = a_row + 32 * it;
            float4 v = ar[it];
            As2[buf][a_kq * 2 + 0][row] = make_float2(v.x * v.x, v.y * v.y);
            As2[buf][a_kq * 2 + 1][row] = make_float2(v.z * v.z, v.w * v.w);
        }
        float4 v0 = br0, v1 = br1;
        if (FUSE_SIGMOID) {
            v0.x = 1.0f / (1.0f + __expf(-v0.x));
            v0.y = 1.0f / (1.0f + __expf(-v0.y));
            v0.z = 1.0f / (1.0f + __expf(-v0.z));
            v0.w = 1.0f / (1.0f + __expf(-v0.w));
            v1.x = 1.0f / (1.0f + __expf(-v1.x));
            v1.y = 1.0f / (1.0f + __expf(-v1.y));
            v1.z = 1.0f / (1.0f + __expf(-v1.z));
            v1.w = 1.0f / (1.0f + __expf(-v1.w));
        }
        Bs2[buf][b_kp][b_nq * 4 + 0] = make_float2(v0.x, v1.x);
        Bs2[buf][b_kp][b_nq * 4 + 1] = make_float2(v0.y, v1.y);
        Bs2[buf][b_kp][b_nq * 4 + 2] = make_float2(v0.z, v1.z);
        Bs2[buf][b_kp][b_nq * 4 + 3] = make_float2(v0.w, v1.w);
    };

    v8f acc[2][2] = {};

    // prologue: stage K-block 0
    load_global(0);
    store_lds(0);
    __syncthreads();

    int cur = 0;
    for (int kb = 0; kb < DIM; kb += BK) {
        const bool has_next = (kb + BK) < DIM;   // uniform
        if (has_next) load_global(kb + BK);

        // advisory GL2 warm for kb+2 (global_prefetch_b8; L2 easily holds both
        // operands, so this mostly helps the first sweep)
        if (kb + 2 * BK < DIM) {
            __builtin_prefetch(X + (size_t)(m0 + a_row) * DIM + (kb + 2 * BK) + a_kq * 4, 0, 1);
            __builtin_prefetch(Bsrc + (size_t)(kb + 2 * BK + 2 * b_kp) * DIM + n0 + b_nq * 4, 0, 1);
        }

        // compute current buffer: 8 K-steps x (2x2) WMMA
#pragma unroll
        for (int kk = 0; kk < BK; kk += 4) {
            const int k2 = (kk >> 1) + half;     // lane-half selects K pair
            v2f afr[2], bfr[2];
#pragma unroll
            for (int i = 0; i < 2; ++i) {
                float2 av = As2[cur][k2][wm + i * 16 + lm];
                afr[i].x = av.x;
                afr[i].y = av.y;
            }
#pragma unroll
            for (int j = 0; j < 2; ++j) {
                float2 bv = Bs2[cur][k2][wn + j * 16 + lm];
                bfr[j].x = bv.x;
                bfr[j].y = bv.y;
            }
#pragma unroll
            for (int i = 0; i < 2; ++i)
#pragma unroll
                for (int j = 0; j < 2; ++j)
                    acc[i][j] = __builtin_amdgcn_wmma_f32_16x16x4_f32(
                        false, afr[i], false, bfr[j],
                        (short)0, acc[i][j], false, false);
        }

        if (has_next) store_lds(cur ^ 1);
        __syncthreads();
        cur ^= 1;
    }

    // C/D layout (ISA 7.12.2): VGPR v -> M = v + 8*half, N = lane%16
#pragma unroll
    for (int i = 0; i < 2; ++i)
#pragma unroll
        for (int j = 0; j < 2; ++j)
#pragma unroll
            for (int v = 0; v < 8; ++v) {
                int m = m0 + wm + i * 16 + v + 8 * half;
                int n = n0 + wn + j * 16 + lm;
                out[(size_t)m * OUT_COLS + INTER_OFF + n] = acc[i][j][v];
            }
}

// ---------------------------------------------------------------------------
// Host launcher
// ---------------------------------------------------------------------------
extern "C" void kernel_launch(void* const* d_in, const int* in_sizes, int n_in,
                              void* d_out, int out_size, void* d_ws, size_t ws_size,
                              hipStream_t stream) {
    const float* x = (const float*)d_in[0];   // (4096, 2048) f32
    const float* W = (const float*)d_in[1];   // (2048, 2048) f32
    float* out = (float*)d_out;               // (4096, 26624) f32

    // Kernel A: basis
    {
        int total  = BATCH * DIM;             // 8,388,608
        int blocks = total / 256;             // 32768
        kan_basis<<<blocks, 256, 0, stream>>>(x, out);
    }

    // Kernel B + C: sigmoid into workspace if it fits, else fuse into GEMM
    const size_t sig_bytes = (size_t)DIM * DIM * sizeof(float);   // 16 MB
    dim3 grid(DIM / BN, BATCH / BM);          // (32, 32)
    if (ws_size >= sig_bytes) {
        float* S = (float*)d_ws;
        int n = DIM * DIM;
        kan_sigmoid<<<(n + 255) / 256, 256, 0, stream>>>(W, S, n);
        kan_gemm<false><<<grid, 256, 0, stream>>>(x, S, out);
    } else {
        kan_gemm<true><<<grid, 256, 0, stream>>>(x, W, out);
    }
}